// ResidualCodingModel_3204045603537
// MI455X (gfx1250) — compile-verified
//
#include <hip/hip_runtime.h>
#include <hip/hip_bf16.h>

typedef float v2f __attribute__((ext_vector_type(2)));
typedef float v8f __attribute__((ext_vector_type(8)));

#define Bsz 1024
#define Lsz 2048
#define Hsz 64
#define DFF 128
#define Vsz 64
#define NT  (Lsz - 2)   // 2046 scan steps
#define MLD 68          // padded stride for 64-wide rows
#define KLD 68
#define TLD 18          // padded stride for 16-wide transposed rows (even -> b64-aligned)

// ---------------- WMMA helpers (fp32 16x16x4, CDNA5) ----------------

__device__ __forceinline__ v8f wmma4(v2f a, v2f b, v8f c) {
  // D = A(16x4,f32) * B(4x16,f32) + C(16x16,f32)
  return __builtin_amdgcn_wmma_f32_16x16x4_f32(
      /*neg_a=*/false, a, /*neg_b=*/false, b,
      /*c_mod=*/(short)0, c, /*reuse_a=*/false, /*reuse_b=*/false);
}

// 8-byte-aligned LDS pair load (indices are even by construction) -> ds_load_b64
__device__ __forceinline__ v2f ld2(const float* p, int idx) {
  return *(const v2f*)(p + idx);
}

// A-matrix f32 16x4 from row-major src: lane&15 = row m, (lane>>4)*2 = K pair (contiguous)
__device__ __forceinline__ v2f lda(const float* p, int ld, int r0, int c0, int lane) {
  int m = lane & 15, k = (lane >> 4) * 2;
  return ld2(p, (r0 + m) * ld + c0 + k);
}
// C/D f32 16x16 accumulator layout: vgpr v -> row r0 + v + 8*(lane>>4), col c0 + (lane&15)
__device__ __forceinline__ v8f ldc(const float* p, int ld, int r0, int c0, int lane) {
  int half = lane >> 4, n = lane & 15;
  v8f c;
#pragma unroll
  for (int v = 0; v < 8; ++v) c[v] = p[(r0 + v + half * 8) * ld + c0 + n];
  return c;
}
__device__ __forceinline__ void stc(float* p, int ld, int r0, int c0, int lane, v8f c) {
  int half = lane >> 4, n = lane & 15;
#pragma unroll
  for (int v = 0; v < 8; ++v) p[(r0 + v + half * 8) * ld + c0 + n] = c[v];
}

// ---------------- Kernel 1: collapse token pipeline into tables ----------------
// T[v] = LN(e_v + MLP(e_v)) ;  P[v] = T[v] @ pred_w + pred_b
__global__ void build_tables(const float* __restrict__ embed_w,
                             const float* __restrict__ ff_w1, const float* __restrict__ ff_b1,
                             const float* __restrict__ ff_w2, const float* __restrict__ ff_b2,
                             const float* __restrict__ ln_g,  const float* __restrict__ ln_b,
                             const float* __restrict__ pred_w, const float* __restrict__ pred_b,
                             float* __restrict__ Tg, float* __restrict__ Pg) {
  int v = blockIdx.x, i = threadIdx.x;
  __shared__ float e[Hsz], hid[DFF], x[Hsz], tv[Hsz], red[2];

  float ei = embed_w[v * Hsz + i];
  e[i] = ei;
  __syncthreads();

#pragma unroll
  for (int c = 0; c < 2; ++c) {
    int d = i + c * Hsz;
    float acc = ff_b1[d];
    for (int j = 0; j < Hsz; ++j) acc += e[j] * ff_w1[j * DFF + d];
    hid[d] = fmaxf(acc, 0.0f);
  }
  __syncthreads();

  float ff = ff_b2[i];
  for (int d = 0; d < DFF; ++d) ff += hid[d] * ff_w2[d * Hsz + i];
  float xi = ei + ff;
  x[i] = xi;
  __syncthreads();

  if (i == 0) { float s = 0.f; for (int j = 0; j < Hsz; ++j) s += x[j]; red[0] = s * (1.0f / Hsz); }
  __syncthreads();
  float mu = red[0];
  if (i == 0) { float s = 0.f; for (int j = 0; j < Hsz; ++j) { float d = x[j] - mu; s += d * d; } red[1] = s * (1.0f / Hsz); }
  __syncthreads();

  float ti = (xi - mu) * rsqrtf(red[1] + 1e-5f) * ln_g[i] + ln_b[i];
  tv[i] = ti;
  Tg[v * Hsz + i] = ti;
  __syncthreads();

  float p = pred_b[i];
  for (int j = 0; j < Hsz; ++j) p += tv[j] * pred_w[j * Hsz + i];
  Pg[v * Hsz + i] = p;
}

// ---------------- Kernel 2: chunked delta-rule scan, LDS-resident M ----------------
__global__ void __launch_bounds__(128, 1)
scan_kernel(const int* __restrict__ seq,
            const float* __restrict__ Tg, const float* __restrict__ Pg,
            const float* __restrict__ read_w, const float* __restrict__ read_b,
            const float* __restrict__ out_w,  const float* __restrict__ out_b,
            float* __restrict__ out) {
  int b = blockIdx.x, tid = threadIdx.x;
  int lane = tid & 31, wv = tid >> 5;

  __shared__ float Ml[Hsz * MLD];       // fast-weight M, row-major, padded
  __shared__ float Tl[Vsz * Hsz];       // token table (L2 -> LDS once)
  __shared__ float Kc[16 * KLD];        // chunk keys, row-major (t, j)
  __shared__ float Kt[Hsz * TLD];       // chunk keys, transposed (j, t)  -> b64 WMMA operands
  __shared__ float Rc[16 * KLD];        // chunk residual targets
  __shared__ float VPl[16 * KLD];       // K @ M0^T
  __shared__ float DVt[Hsz * TLD];      // solved dv, transposed (j, t)   -> b64 WMMA operands
  __shared__ float Gl[16 * 16];         // Gram K K^T
  __shared__ float dninv[16];           // 1 / (||k||^2 + eps)
  __shared__ int   tokA[16], tokB[16];
  __shared__ float fin[2 * Hsz];

  for (int idx = tid; idx < Hsz * MLD; idx += 128) Ml[idx] = 0.0f;
  for (int idx = tid; idx < Vsz * Hsz; idx += 128) Tl[idx] = Tg[idx];
  __syncthreads();

  const int* sb = seq + (long)b * Lsz;

  for (int t0 = 0; t0 < NT; t0 += 16) {
    // ---- stage chunk tokens; prefetch next window ----
    if (tid < 16) {
      int tt = t0 + tid;
      tokA[tid] = (tt < NT) ? sb[tt] : -1;
      tokB[tid] = (tt < NT) ? sb[tt + 1] : -1;
      __builtin_prefetch(sb + tt + 16, 0, 0);
    }
    __syncthreads();

    // ---- gather K, R rows (zero-pad partial chunk; padded rows yield dv=0) ----
    for (int idx = tid; idx < 16 * Hsz; idx += 128) {
      int t = idx >> 6, j = idx & 63;
      int a = tokA[t];
      float kv = 0.0f, rv = 0.0f;
      if (a >= 0) {
        kv = Tl[a * Hsz + j];
        rv = Tl[tokB[t] * Hsz + j] - Pg[a * Hsz + j];
      }
      Kc[t * KLD + j] = kv;
      Kt[j * TLD + t] = kv;
      Rc[t * KLD + j] = rv;
    }
    __syncthreads();

    // ---- reciprocal denominators ----
    if (tid < 16) {
      float s = 1e-6f;
      for (int j = 0; j < Hsz; ++j) { float k = Kc[tid * KLD + j]; s += k * k; }
      dninv[tid] = 1.0f / s;
    }

    // ---- G = K K^T (wave 0): B[k][s] = Kc[s][jc*4+k] = Kc row s, contiguous pair ----
    if (wv == 0) {
      v8f g = {};
#pragma unroll
      for (int jc = 0; jc < 16; ++jc) {
        v2f a = lda(Kc, KLD, 0, jc * 4, lane);
        int n = lane & 15, k = (lane >> 4) * 2;
        v2f bb = ld2(Kc, n * KLD + jc * 4 + k);
        g = wmma4(a, bb, g);
      }
      stc(Gl, 16, 0, 0, lane, g);
    }

    // ---- VP = K @ M0^T, wave w computes column block [16w, 16w+16) ----
    {
      int i0 = wv * 16;
      v8f vp = {};
#pragma unroll
      for (int jc = 0; jc < 16; ++jc) {
        v2f a = lda(Kc, KLD, 0, jc * 4, lane);
        int n = lane & 15, k = (lane >> 4) * 2;     // B[k][n] = M[i0+n][jc*4+k], contiguous pair
        v2f bb = ld2(Ml, (i0 + n) * MLD + jc * 4 + k);
        vp = wmma4(a, bb, vp);
      }
      stc(VPl, KLD, 0, i0, lane, vp);
    }
    __syncthreads();

    // ---- forward substitution: dv_t = r_t - (VP_t + sum_{s<t} G[t][s] dv_s) * dninv_t ----
    if (tid < Hsz) {
      int j = tid;
      float dvloc[16];
#pragma unroll
      for (int t = 0; t < 16; ++t) {
        float acc = VPl[t * KLD + j];
        for (int s = 0; s < t; ++s) acc += Gl[t * 16 + s] * dvloc[s];
        float dv = Rc[t * KLD + j] - acc * dninv[t];
        dvloc[t] = dv;
        DVt[j * TLD + t] = dv;
      }
    }
    __syncthreads();

    // ---- M += DV^T @ K  (wave w owns row block [16w, 16w+16)) ----
    {
      int i0 = wv * 16;
#pragma unroll
      for (int jt = 0; jt < 4; ++jt) {
        int j0 = jt * 16;
        v8f c = ldc(Ml, MLD, i0, j0, lane);
#pragma unroll
        for (int tc = 0; tc < 4; ++tc) {
          int m = lane & 15, k = (lane >> 4) * 2;
          v2f a  = ld2(DVt, (i0 + m) * TLD + tc * 4 + k);  // A[m][k] = dv_{tc*4+k}[i0+m]
          v2f bb = ld2(Kt,  (j0 + (lane & 15)) * TLD + tc * 4 + k); // B[k][n] = K[tc*4+k][j0+n]
          c = wmma4(a, bb, c);
        }
        stc(Ml, MLD, i0, j0, lane, c);
      }
    }
    __syncthreads();
  }

  // ---- readout: ctx = M q ; out = (ctx@read_w + read_b)@out_w + out_b ----
  int tok = sb[Lsz - 1];
  if (tid < Hsz) {
    float ctx = 0.0f;
    for (int j = 0; j < Hsz; ++j) ctx += Ml[tid * MLD + j] * Tl[tok * Hsz + j];
    fin[tid] = ctx;
  }
  __syncthreads();
  if (tid < Hsz) {
    float y = read_b[tid];
    for (int j = 0; j < Hsz; ++j) y += fin[j] * read_w[j * Hsz + tid];
    fin[Hsz + tid] = y;
  }
  __syncthreads();
  if (tid < Hsz) {
    float o = out_b[tid];
    for (int j = 0; j < Hsz; ++j) o += fin[Hsz + j] * out_w[j * Vsz + tid];
    out[(long)b * Vsz + tid] = o;
  }
}

// ---------------- launch ----------------
extern "C" void kernel_launch(void* const* d_in, const int* in_sizes, int n_in,
                              void* d_out, int out_size, void* d_ws, size_t ws_size,
                              hipStream_t stream) {
  const int*   seq     = (const int*)d_in[0];
  const float* embed_w = (const float*)d_in[1];
  const float* ff_w1   = (const float*)d_in[2];
  const float* ff_b1   = (const float*)d_in[3];
  const float* ff_w2   = (const float*)d_in[4];
  const float* ff_b2   = (const float*)d_in[5];
  const float* ln_g    = (const float*)d_in[6];
  const float* ln_b    = (const float*)d_in[7];
  const float* pred_w  = (const float*)d_in[8];
  const float* pred_b  = (const float*)d_in[9];
  const float* read_w  = (const float*)d_in[10];
  const float* read_b  = (const float*)d_in[11];
  const float* out_w   = (const float*)d_in[12];
  const float* out_b   = (const float*)d_in[13];

  float* Tg = (float*)d_ws;          // 64*64 floats
  float* Pg = Tg + Vsz * Hsz;        // 64*64 floats

  build_tables<<<Vsz, Hsz, 0, stream>>>(embed_w, ff_w1, ff_b1, ff_w2, ff_b2,
                                        ln_g, ln_b, pred_w, pred_b, Tg, Pg);
  scan_kernel<<<Bsz, 128, 0, stream>>>(seq, Tg, Pg, read_w, read_b, out_w, out_b,
                                       (float*)d_out);
}